// ExtractTsFeatures_86483461472737
// MI455X (gfx1250) — compile-verified
//
#include <hip/hip_runtime.h>
#include <stdint.h>
#include <stddef.h>

// ---------------------------------------------------------------------------
// ExtractTsFeatures for x:(512,1024,32) f32 -> out:(512,32,30) f32
//
// MI455X strategy: memory-bound reduction+selection (64 MiB in, ~2.8us HBM
// floor @23.3TB/s). No matmul structure -> WMMA inapplicable; the CDNA5 paths
// that pay are the Tensor Data Mover (one 128KB descriptor DMA of x[b] into
// the 320KB/WGP LDS, with hardware row padding to a 33-dword stride so the
// column-major bitonic sort is bank-conflict free) and wave32 block shaping
// (lane==feature, wave==t-phase).
// ---------------------------------------------------------------------------

#define TT        1024
#define FF        32
#define ROWP      33            // padded LDS row stride in dwords (TDM pads)
#define NTHREADS  256
#define NGRP      8             // NTHREADS / 32

#define XS_FLOATS   (TT * ROWP)        // 33792
#define RED_FLOATS  (10 * NTHREADS)    // 2560
#define MEAN_FLOATS 32
#define SMEM_FLOATS (XS_FLOATS + RED_FLOATS + MEAN_FLOATS)
#define SMEM_BYTES  (SMEM_FLOATS * 4)  // 145,536 B < 320 KB/WGP

typedef unsigned int u32x4 __attribute__((ext_vector_type(4)));
typedef int          i32x8 __attribute__((ext_vector_type(8)));
typedef int          i32x4 __attribute__((ext_vector_type(4)));

__global__ void ExtractTsFeatures_86483461472737_kernel(const float* __restrict__ x,
                                                        float* __restrict__ out) {
  extern __shared__ float smem[];
  float* xs    = smem;                 // [TT*ROWP] padded tile of x[b]
  float* red   = smem + XS_FLOATS;     // [10*NTHREADS] reduction scratch
  float* meanb = red + RED_FLOATS;     // [32] per-feature means

  const int tid = threadIdx.x;
  const int b   = blockIdx.x;
  const int f   = tid & 31;            // feature lane
  const int g   = tid >> 5;            // t-phase group (== wave id)

  // ---- Phase 0: TDM DMA of x[b] (1024x32 f32, contiguous 128KB) into LDS,
  //      padded +1 dword per 32-dword row (D# pad_interval=4, pad_amount=0).
#if __has_builtin(__builtin_amdgcn_tensor_load_to_lds)
  if (tid == 0) {
    unsigned long long gaddr =
        (unsigned long long)(uintptr_t)(x + (size_t)b * TT * FF);
    unsigned int lds_off = (unsigned int)(uintptr_t)xs; // low 32b = LDS offset

    u32x4 g0;
    g0.x = 1u;                                          // count=1, user mode
    g0.y = lds_off;                                     // lds_addr [63:32]
    g0.z = (unsigned int)(gaddr & 0xFFFFFFFFull);       // global_addr [95:64]
    g0.w = (unsigned int)((gaddr >> 32) & 0x1FFFFFFull) // global_addr [120:96]
         | (2u << 30);                                  // type=2 ("image")

    i32x8 g1;
    g1[0] = 0x01120000;        // data_size=4B, pad_enable=1, pad_interval=32dw
    g1[1] = (int)0x80000000u;  // tensor_dim0 = 32768 (lo16 in [31:16])
    g1[2] = 0x00010000;        // tensor_dim0 hi=0; tensor_dim1 = 1
    g1[3] = (int)0x80000000u;  // tile_dim0 = 32768 elements (1-D tile)
    g1[4] = 0;                 // tile_dim1 = tile_dim2 = 0 (unused)
    g1[5] = 0x8000;            // tensor_dim0_stride = 32768
    g1[6] = 0;
    g1[7] = 0;

    i32x4 zz4 = {0, 0, 0, 0};              // groups 2/3 unused (<=2D tensor)
    i32x8 zz8 = {0, 0, 0, 0, 0, 0, 0, 0};  // 6th-arg group (clang-23 form)
    __builtin_amdgcn_tensor_load_to_lds(g0, g1, zz4, zz4, zz8, 0);
    __builtin_amdgcn_s_wait_tensorcnt(0);
  }
#else
  for (int i = tid; i < TT * FF; i += NTHREADS) {
    int t = i >> 5, ff = i & 31;
    xs[t * ROWP + ff] = x[(size_t)b * TT * FF + i];
  }
#endif
  __syncthreads();

  // ---- Phase 1: raw-moment / diff partials. Thread (g,f) covers t=g,g+8,...
  {
    float s = 0.f, s2 = 0.f, cp = 0.f;
    float mn = 3.402823466e38f, mx = -3.402823466e38f, ma = 0.f;
    float dsum = 0.f, dabs = 0.f, dsq = 0.f;
    for (int t = g; t < TT; t += NGRP) {
      float v = xs[t * ROWP + f];
      s  += v;
      s2 += v * v;
      mn = fminf(mn, v);
      mx = fmaxf(mx, v);
      ma = fmaxf(ma, fabsf(v));
      cp += (v > 0.f) ? 1.f : 0.f;
      if (t >= 1 && t <= TT - 2) {          // x_diff = x[1:-1] - x[2:]
        float d = v - xs[(t + 1) * ROWP + f];
        dsum += d;
        dabs += fabsf(d);
        dsq  += d * d;
      }
    }
    red[0 * NTHREADS + tid] = s;
    red[1 * NTHREADS + tid] = s2;
    red[2 * NTHREADS + tid] = mn;
    red[3 * NTHREADS + tid] = mx;
    red[4 * NTHREADS + tid] = ma;
    red[5 * NTHREADS + tid] = cp;
    red[6 * NTHREADS + tid] = dsum;
    red[7 * NTHREADS + tid] = dabs;
    red[8 * NTHREADS + tid] = dsq;
  }
  __syncthreads();

  if (tid < 32) {
    float S = 0.f, S2 = 0.f, CP = 0.f, DS = 0.f, DA = 0.f, DQ = 0.f;
    float MN = 3.402823466e38f, MX = -3.402823466e38f, MA = 0.f;
    for (int gg = 0; gg < NGRP; ++gg) {
      int i = gg * 32 + tid;
      S  += red[0 * NTHREADS + i];
      S2 += red[1 * NTHREADS + i];
      MN  = fminf(MN, red[2 * NTHREADS + i]);
      MX  = fmaxf(MX, red[3 * NTHREADS + i]);
      MA  = fmaxf(MA, red[4 * NTHREADS + i]);
      CP += red[5 * NTHREADS + i];
      DS += red[6 * NTHREADS + i];
      DA += red[7 * NTHREADS + i];
      DQ += red[8 * NTHREADS + i];
    }
    float mean = S / 1024.f;
    meanb[tid] = mean;
    float msq = S2 / 1024.f;
    float* o = out + ((size_t)b * 32 + tid) * 30;
    o[0]  = mean;
    o[1]  = MN;
    o[2]  = MX;
    o[3]  = (msq == 0.f) ? 0.f : sqrtf(msq);   // _safe_sqrt
    o[8]  = DS / 1022.f;                        // mean(x_diff), N=T-2
    o[9]  = DS;
    o[10] = DA / 1022.f;
    // tb samples (pre-sort reads): t = 0,256,512,767,1023
    o[14] = xs[0 * ROWP + tid];
    o[15] = xs[256 * ROWP + tid];
    o[16] = xs[512 * ROWP + tid];
    o[17] = xs[767 * ROWP + tid];
    o[18] = xs[1023 * ROWP + tid];
    o[19] = S2;
    o[20] = MA;
    o[21] = DA;
    o[22] = (DQ == 0.f) ? 0.f : sqrtf(DQ);
    o[23] = CP;
  }
  __syncthreads();

  // ---- Phase 2: exact centered moments + threshold counts.
  {
    float m  = meanb[f];
    float t0 = xs[0 * ROWP + f];
    float t1 = xs[256 * ROWP + f];
    float t2 = xs[512 * ROWP + f];
    float t3 = xs[767 * ROWP + f];
    float t4 = xs[1023 * ROWP + f];
    float p2 = 0.f, p3 = 0.f, p4 = 0.f, cg = 0.f;
    float c0 = 0.f, c1 = 0.f, c2 = 0.f, c3 = 0.f, c4 = 0.f;
    for (int t = g; t < TT; t += NGRP) {
      float v  = xs[t * ROWP + f];
      float d  = v - m;
      float dd = d * d;
      p2 += dd;
      p3 += dd * d;
      p4 += dd * dd;
      cg += (d > 0.f) ? 1.f : 0.f;
      c0 += (v > t0) ? 1.f : 0.f;
      c1 += (v > t1) ? 1.f : 0.f;
      c2 += (v > t2) ? 1.f : 0.f;
      c3 += (v > t3) ? 1.f : 0.f;
      c4 += (v > t4) ? 1.f : 0.f;
    }
    red[0 * NTHREADS + tid] = p2;
    red[1 * NTHREADS + tid] = p3;
    red[2 * NTHREADS + tid] = p4;
    red[3 * NTHREADS + tid] = cg;
    red[4 * NTHREADS + tid] = c0;
    red[5 * NTHREADS + tid] = c1;
    red[6 * NTHREADS + tid] = c2;
    red[7 * NTHREADS + tid] = c3;
    red[8 * NTHREADS + tid] = c4;
  }
  __syncthreads();

  if (tid < 32) {
    float A[9] = {0.f, 0.f, 0.f, 0.f, 0.f, 0.f, 0.f, 0.f, 0.f};
    for (int gg = 0; gg < NGRP; ++gg) {
      int i = gg * 32 + tid;
      for (int q = 0; q < 9; ++q) A[q] += red[q * NTHREADS + i];
    }
    float P2 = A[0], P3 = A[1], P4 = A[2];
    float var = P2 / 1024.f;
    float sd  = (var == 0.f) ? 0.f : sqrtf(var);
    float* o  = out + ((size_t)b * 32 + tid) * 30;
    o[4] = var;
    o[5] = sd;
    const float CSK = 1024.f / (1023.f * 1022.f);           // nT/((nT-1)(nT-2))
    o[6] = (sd == 0.f) ? 0.f : CSK * P3 / (sd * sd * sd);   // _dnn(x_sub,std)
    const float ALPHA = (1024.f * 1025.f * 1023.f) / (1022.f * 1021.f);
    const float RIGHT = (3.f * 1023.f * 1023.f) / (1022.f * 1021.f);
    float k22 = P2 * P2;
    o[7]  = ALPHA * ((k22 == 0.f) ? 0.f : P4 / k22) - RIGHT;
    o[24] = A[3];
    o[25] = A[4];
    o[26] = A[5];
    o[27] = A[6];
    o[28] = A[7];
    o[29] = A[8];
  }
  __syncthreads();

  // ---- Phase 3: in-LDS bitonic sort of each of the 32 columns (N=1024) for
  //      exact order statistics. 16384 CEs/stage spread over 256 threads;
  //      lanes within a wave hit consecutive dwords (stride-33 rows) -> no
  //      bank conflicts.
  for (int k = 2; k <= TT; k <<= 1) {
    for (int j = k >> 1; j > 0; j >>= 1) {
      for (int c = 0; c < 64; ++c) {
        int pid = tid + NTHREADS * c;
        int ff  = pid & 31;
        int p   = pid >> 5;
        int i   = ((p & ~(j - 1)) << 1) | (p & (j - 1));
        int l   = i | j;
        bool up = ((i & k) == 0);
        float a  = xs[i * ROWP + ff];
        float bv = xs[l * ROWP + ff];
        float lo = fminf(a, bv);
        float hi = fmaxf(a, bv);
        xs[i * ROWP + ff] = up ? lo : hi;
        xs[l * ROWP + ff] = up ? hi : lo;
      }
      __syncthreads();
    }
  }

  if (tid < 32) {
    float* o = out + ((size_t)b * 32 + tid) * 30;
    o[11] = xs[256 * ROWP + tid];   // q25: round(0.25*1023)=256
    o[12] = xs[512 * ROWP + tid];   // q50: round(511.5)=512 (banker's)
    o[13] = xs[767 * ROWP + tid];   // q75: round(767.25)=767
  }
}

extern "C" void kernel_launch(void* const* d_in, const int* in_sizes, int n_in,
                              void* d_out, int out_size, void* d_ws, size_t ws_size,
                              hipStream_t stream) {
  (void)in_sizes; (void)n_in; (void)d_ws; (void)ws_size; (void)out_size;
  const float* x  = (const float*)d_in[0];
  float*       out = (float*)d_out;
  (void)hipFuncSetAttribute((const void*)ExtractTsFeatures_86483461472737_kernel,
                            hipFuncAttributeMaxDynamicSharedMemorySize, SMEM_BYTES);
  ExtractTsFeatures_86483461472737_kernel<<<512, NTHREADS, SMEM_BYTES, stream>>>(x, out);
}